// MFHawkes_32822140076808
// MI455X (gfx1250) — compile-verified
//
#include <hip/hip_runtime.h>
#include <math.h>

// Problem dims (match reference: B=4, L=2048, K=100000, D=64)
#define BB 4
#define LL 2048
#define DD 64
#define NT (LL / 16)   // 128 row tiles per batch

typedef float v2f __attribute__((ext_vector_type(2)));
typedef float v8f __attribute__((ext_vector_type(8)));

// ---------------- kernel 1: per-event scalars ----------------
__global__ void mfh_prep(const float* __restrict__ times,
                         const float* __restrict__ mask,
                         const int*   __restrict__ ids,
                         const float* __restrict__ u_table,
                         const float* __restrict__ beta,
                         float* __restrict__ t_ws, float* __restrict__ g_ws,
                         float* __restrict__ u_ws, float* __restrict__ w_ws) {
    const int j = blockIdx.x * blockDim.x + threadIdx.x;
    const int b = blockIdx.y;
    if (j >= LL) return;
    const float ab    = fabsf(beta[0]);
    const float t     = times[b * LL + j] * 1e-4f;
    const float tlast = times[b * LL + (LL - 1)] * 1e-4f;
    t_ws[b * LL + j] = t;
    g_ws[b * LL + j] = mask[b * LL + j] * expf(ab * t);        // mask folded in
    u_ws[b * LL + j] = fabsf(u_table[ids[b * LL + j]]);
    w_ws[b * LL + j] = 1.0f - expf(-ab * (tlast - t));
}

// ---------------- kernel 2: gather embedding rows ----------------
__global__ void mfh_gather(const int* __restrict__ ids,
                           const float* __restrict__ emb,
                           float* __restrict__ e_ws) {
    const int n = blockIdx.x * blockDim.x + threadIdx.x;   // over B*L*D
    if (n >= BB * LL * DD) return;
    const int d  = n & (DD - 1);
    const int bj = n >> 6;                                  // b*L + j
    e_ws[n] = emb[(size_t)ids[bj] * DD + d];
}

// ---------------- kernel 3: v_b = sum_j w_j * e_j ----------------
__global__ void mfh_vsum(const float* __restrict__ e_ws,
                         const float* __restrict__ w_ws,
                         float* __restrict__ v_ws) {
    const int b = blockIdx.x;
    const int d = threadIdx.x;                              // 64 threads
    float acc = 0.0f;
    for (int j = 0; j < LL; ++j)
        acc = fmaf(w_ws[b * LL + j], e_ws[((size_t)(b * LL + j)) * DD + d], acc);
    v_ws[b * DD + d] = acc;
}

// ---------------- kernel 4: WMMA triangular weighted Gram row-sums ----------------
__global__ void __launch_bounds__(32)
mfh_tile(const float* __restrict__ times,
         const float* __restrict__ beta,
         const float* __restrict__ e_ws,
         const float* __restrict__ t_ws,
         const float* __restrict__ g_ws,
         const float* __restrict__ u_ws,
         const float* __restrict__ v_ws,
         float* __restrict__ partial) {
    const int I    = blockIdx.x;          // row tile
    const int b    = blockIdx.y;          // batch
    const int lane = threadIdx.x;         // 0..31, one wave
    const int m    = lane & 15;
    const int h    = lane >> 4;           // half-wave selects K-pair / row group
    const int rowbase = I * 16;

    const float ab      = fabsf(beta[0]);
    const float horizon = (times[b * LL + (LL - 1)] - times[b * LL + 1]) * 1e-4f;

    // A tile: rows rowbase..rowbase+15, K=0..63.
    // f32 16x16x4 A layout: lane m holds row m; VGPR0/1 = K {2h, 2h+1} of the step's K-quad.
    v2f a[16], vv[16];
    const float* arow = e_ws + ((size_t)(b * LL + rowbase + m)) * DD;
    const float* vb   = v_ws + b * DD;
#pragma unroll
    for (int s = 0; s < 16; ++s) {
        a[s]  = *(const v2f*)(arow + s * 4 + 2 * h);
        vv[s] = *(const v2f*)(vb   + s * 4 + 2 * h);
    }

    float acc[8] = {0.f, 0.f, 0.f, 0.f, 0.f, 0.f, 0.f, 0.f};

    // ---- off-diagonal tiles: mask always passes -> pure FMA accumulate ----
    for (int J = 0; J < I; ++J) {
        const int colbase = J * 16;
        const float* brow = e_ws + ((size_t)(b * LL + colbase + m)) * DD;
        v2f bbv[16];
#pragma unroll
        for (int s = 0; s < 16; ++s)
            bbv[s] = *(const v2f*)(brow + s * 4 + 2 * h);

        v8f c = {0.f, 0.f, 0.f, 0.f, 0.f, 0.f, 0.f, 0.f};
#pragma unroll
        for (int s = 0; s < 16; ++s)
            c = __builtin_amdgcn_wmma_f32_16x16x4_f32(
                    false, a[s], false, bbv[s], (short)0, c, false, false);

        // C/D layout: c[v] @ lane = alpha[rowbase + v + 8h][colbase + m]
        const float gc = g_ws[b * LL + colbase + m];
#pragma unroll
        for (int v = 0; v < 8; ++v)
            acc[v] = fmaf(c[v], gc, acc[v]);
    }

    // ---- diagonal tile (J == I): strict-lower mask col < row ----
    {
        const int colbase = rowbase;
        const float* brow = e_ws + ((size_t)(b * LL + colbase + m)) * DD;
        v2f bbv[16];
#pragma unroll
        for (int s = 0; s < 16; ++s)
            bbv[s] = *(const v2f*)(brow + s * 4 + 2 * h);

        v8f c = {0.f, 0.f, 0.f, 0.f, 0.f, 0.f, 0.f, 0.f};
#pragma unroll
        for (int s = 0; s < 16; ++s)
            c = __builtin_amdgcn_wmma_f32_16x16x4_f32(
                    false, a[s], false, bbv[s], (short)0, c, false, false);

        const int   col = colbase + m;
        const float gc  = g_ws[b * LL + col];
#pragma unroll
        for (int v = 0; v < 8; ++v) {
            const int row = rowbase + v + 8 * h;
            acc[v] += (col < row) ? c[v] * gc : 0.0f;   // tril, k = -1
        }
    }

    // Row-sum: reduce across the 16 lanes of each half-wave.
#pragma unroll
    for (int v = 0; v < 8; ++v)
        for (int off = 1; off < 16; off <<= 1)
            acc[v] += __shfl_xor(acc[v], off, 16);

    // e_i . v_b using the A registers (row m split across the two halves).
    float p = 0.0f;
#pragma unroll
    for (int s = 0; s < 16; ++s)
        p += a[s].x * vv[s].x + a[s].y * vv[s].y;
    const float dot = p + __shfl_xor(p, 16, 32);        // all lanes: dot of row (lane&15)

    // Lanes with (m>>3)==h own row rowbase+m: have both S1 (=acc[m&7]) and dot.
    float ll = 0.0f;
    if (((lane >> 3) & 1) == h) {
        const int   r  = rowbase + m;
        const float ti = t_ws[b * LL + r];
        const float ui = u_ws[b * LL + r];
        const float S1 = acc[m & 7];
        const float di = ab * expf(-ab * ti);
        const float lam = fabsf(di * S1 + ui) + 1e-6f;
        ll = -logf(lam) + ui * horizon + dot;           // = -log(lam) - negative_integral
    }
    for (int off = 1; off < 32; off <<= 1)
        ll += __shfl_xor(ll, off, 32);
    if (lane == 0) partial[b * NT + I] = ll;
}

// ---------------- kernel 5: deterministic tree reduction ----------------
__global__ void mfh_reduce(const float* __restrict__ partial, float* __restrict__ out) {
    __shared__ float sm[NT];
    const int b = blockIdx.x;
    sm[threadIdx.x] = partial[b * NT + threadIdx.x];
    __syncthreads();
    for (int s = NT / 2; s > 0; s >>= 1) {
        if (threadIdx.x < s) sm[threadIdx.x] += sm[threadIdx.x + s];
        __syncthreads();
    }
    if (threadIdx.x == 0) out[b] = sm[0];
}

extern "C" void kernel_launch(void* const* d_in, const int* in_sizes, int n_in,
                              void* d_out, int out_size, void* d_ws, size_t ws_size,
                              hipStream_t stream) {
    (void)in_sizes; (void)n_in; (void)out_size; (void)ws_size;
    const int*   ids     = (const int*)  d_in[0];
    const float* times   = (const float*)d_in[1];
    const float* mask    = (const float*)d_in[2];
    const float* emb     = (const float*)d_in[3];
    const float* u_table = (const float*)d_in[4];
    const float* beta    = (const float*)d_in[5];
    float*       out     = (float*)d_out;

    float* ws = (float*)d_ws;
    // Workspace layout (floats): total ~557,824 floats (~2.23 MB)
    float* e_ws = ws;                          // B*L*D = 524288
    float* t_ws = e_ws + (size_t)BB * LL * DD; // B*L
    float* g_ws = t_ws + BB * LL;              // B*L
    float* u_ws = g_ws + BB * LL;              // B*L
    float* w_ws = u_ws + BB * LL;              // B*L
    float* v_ws = w_ws + BB * LL;              // B*D
    float* p_ws = v_ws + BB * DD;              // B*NT

    mfh_prep<<<dim3(LL / 256, BB), 256, 0, stream>>>(times, mask, ids, u_table, beta,
                                                     t_ws, g_ws, u_ws, w_ws);
    mfh_gather<<<(BB * LL * DD) / 256, 256, 0, stream>>>(ids, emb, e_ws);
    mfh_vsum<<<BB, DD, 0, stream>>>(e_ws, w_ws, v_ws);
    mfh_tile<<<dim3(NT, BB), 32, 0, stream>>>(times, beta, e_ws, t_ws, g_ws, u_ws,
                                              v_ws, p_ws);
    mfh_reduce<<<BB, NT, 0, stream>>>(p_ws, out);
}